// BinaryDiceLoss_blobPunish_6047313953508
// MI455X (gfx1250) — compile-verified
//
#include <hip/hip_runtime.h>
#include <hip/hip_bf16.h>
#include <cstdint>
#include <cstddef>

// ---------------------------------------------------------------------------
// Problem geometry
// ---------------------------------------------------------------------------
#define IMG_H 512
#define IMG_W 512
#define IMG_N (IMG_H * IMG_W)     // 262144
#define BATCH 16
#define NTOT  (BATCH * IMG_N)     // 4194304

// CCL LDS tiling: 64x64 output tile, 5 propagation iterations per launch.
#define TILE  64
#define KIT   5
#define REG   (TILE + 2 * KIT)    // 74 loaded region
#define DIM   (REG + 2)           // 76 incl. zero ring
#define PITCH (DIM + 1)           // 77 LDS row pitch
#define CCL_LAUNCHES 40           // 40 * 5 = 200 iterations

// ---------------------------------------------------------------------------
// Workspace layout (bytes). Total ~40 MiB.
// ---------------------------------------------------------------------------
constexpr size_t BUF_BYTES = (size_t)NTOT * 4;        // 16 MiB label buffer
constexpr size_t MAP_BYTES = 4u * 1024 * 1024 + 256;  // 4,194,560 >= NTOT+1, 16B mult
constexpr size_t OFF_BUFA  = 0;
constexpr size_t OFF_BUFB  = BUF_BYTES;
constexpr size_t OFF_MAPP  = 2 * BUF_BYTES;
constexpr size_t OFF_MAPT  = OFF_MAPP + MAP_BYTES;
constexpr size_t OFF_SCAL  = OFF_MAPT + MAP_BYTES;    // u32: keyP,keyT,cntP,cntT
constexpr size_t OFF_SUMS  = OFF_SCAL + 256;          // 32 floats: per-image num/den

typedef float v2f __attribute__((ext_vector_type(2)));
typedef float v8f __attribute__((ext_vector_type(8)));

// Order-preserving float->uint key (total order matches float compare).
__device__ __forceinline__ unsigned fkey(float x) {
  unsigned b = __float_as_uint(x);
  return b ^ (unsigned)(((int)b >> 31) | (int)0x80000000u);
}
__device__ __forceinline__ float funkey(unsigned k) {
  unsigned b = (k & 0x80000000u) ? (k ^ 0x80000000u) : ~k;
  return __uint_as_float(b);
}

// ---------------------------------------------------------------------------
// 0) Zero the presence maps + scalar accumulators (re-run every launch).
// ---------------------------------------------------------------------------
__global__ void init_ws_kernel(unsigned* __restrict__ z, int nwords) {
  for (int i = blockIdx.x * blockDim.x + threadIdx.x; i < nwords;
       i += gridDim.x * blockDim.x)
    z[i] = 0u;
}

// ---------------------------------------------------------------------------
// 1) Global max of predict and target (uint atomicMax on ordered keys).
// ---------------------------------------------------------------------------
__global__ __launch_bounds__(256) void max_kernel(const float* __restrict__ p,
                                                  const float* __restrict__ t,
                                                  unsigned* __restrict__ keys, int n) {
  float mp = -INFINITY, mt = -INFINITY;
  for (int i = blockIdx.x * blockDim.x + threadIdx.x; i < n;
       i += gridDim.x * blockDim.x) {
    mp = fmaxf(mp, p[i]);
    mt = fmaxf(mt, t[i]);
  }
  for (int off = 16; off > 0; off >>= 1) {
    mp = fmaxf(mp, __shfl_xor(mp, off, 32));
    mt = fmaxf(mt, __shfl_xor(mt, off, 32));
  }
  __shared__ float sp[8], st[8];
  int wave = threadIdx.x >> 5, lane = threadIdx.x & 31;
  if (lane == 0) { sp[wave] = mp; st[wave] = mt; }
  __syncthreads();
  if (threadIdx.x == 0) {
    for (int w = 1; w < 8; ++w) { mp = fmaxf(mp, sp[w]); mt = fmaxf(mt, st[w]); }
    atomicMax(&keys[0], fkey(mp));
    atomicMax(&keys[1], fkey(mt));
  }
}

// ---------------------------------------------------------------------------
// 2) seed = mask ? (flat_index+1) : 0   (exact in f32: indices < 2^24)
// ---------------------------------------------------------------------------
__global__ __launch_bounds__(256) void seed_kernel(const float* __restrict__ x,
                                                   const unsigned* __restrict__ keys,
                                                   int keyIdx,
                                                   float* __restrict__ seed, int n) {
  int i = blockIdx.x * blockDim.x + threadIdx.x;
  if (i >= n) return;
  float thr = funkey(keys[keyIdx]) * 0.5f;
  seed[i] = (x[i] > thr) ? (float)(i + 1) : 0.0f;
}

// ---------------------------------------------------------------------------
// 3) CCL propagation: 5 iterations of  new = (old!=0) ? max3x3(old) : 0
//    per launch, done in LDS with a K=5 halo. Halo is filled with CDNA5
//    async global->LDS loads (ASYNCcnt path) and s_wait_asynccnt.
// ---------------------------------------------------------------------------
__global__ __launch_bounds__(256) void ccl_kernel(const float* __restrict__ in,
                                                  float* __restrict__ out) {
  __shared__ float A[DIM][PITCH];
  __shared__ float B[DIM][PITCH];
  const int tid = threadIdx.x;
  const int gx0 = (int)blockIdx.x * TILE - KIT;  // global x of REG column 0
  const int gy0 = (int)blockIdx.y * TILE - KIT;
  const float* __restrict__ imgIn  = in  + (size_t)blockIdx.z * IMG_N;
  float* __restrict__       imgOut = out + (size_t)blockIdx.z * IMG_N;

  // Fill LDS: ring + out-of-image cells = 0 (plain DS stores), in-image
  // interior via async global->LDS loads (disjoint cells, so no ordering
  // hazard between the DS stores and the async writes).
  for (int idx = tid; idx < DIM * DIM; idx += 256) {
    int r = idx / DIM, c = idx % DIM;
    B[r][c] = 0.0f;
    bool ring = (r == 0) | (r == DIM - 1) | (c == 0) | (c == DIM - 1);
    int gy = gy0 + r - 1, gx = gx0 + c - 1;
    bool inb = (!ring) & (gy >= 0) & (gy < IMG_H) & (gx >= 0) & (gx < IMG_W);
    if (inb) {
      unsigned lds_addr = (unsigned)(uintptr_t)&A[r][c];
      const float* gp = imgIn + gy * IMG_W + gx;
      asm volatile("global_load_async_to_lds_b32 %0, %1, off"
                   :: "v"(lds_addr), "v"(gp)
                   : "memory");
    } else {
      A[r][c] = 0.0f;
    }
  }
  asm volatile("s_wait_asynccnt 0" ::: "memory");
  __syncthreads();

  float (*src)[PITCH] = A;
  float (*dst)[PITCH] = B;
#pragma unroll 1
  for (int it = 0; it < KIT; ++it) {
    for (int idx = tid; idx < REG * REG; idx += 256) {
      int r = idx / REG + 1, c = idx % REG + 1;
      float center = src[r][c];
      float m = center;
      m = fmaxf(m, src[r - 1][c - 1]);
      m = fmaxf(m, src[r - 1][c    ]);
      m = fmaxf(m, src[r - 1][c + 1]);
      m = fmaxf(m, src[r    ][c - 1]);
      m = fmaxf(m, src[r    ][c + 1]);
      m = fmaxf(m, src[r + 1][c - 1]);
      m = fmaxf(m, src[r + 1][c    ]);
      m = fmaxf(m, src[r + 1][c + 1]);
      dst[r][c] = (center != 0.0f) ? m : 0.0f;
    }
    __syncthreads();
    float (*tmp)[PITCH] = src; src = dst; dst = tmp;
  }

  // Central 64x64 of `src` is exact after KIT iterations.
  for (int idx = tid; idx < TILE * TILE; idx += 256) {
    int y = idx / TILE, x = idx % TILE;
    int gy = (int)blockIdx.y * TILE + y, gx = (int)blockIdx.x * TILE + x;
    imgOut[gy * IMG_W + gx] = src[1 + KIT + y][1 + KIT + x];
  }
}

// ---------------------------------------------------------------------------
// 4) Mark presence bytemap: labels are exact integers in [0, NTOT].
// ---------------------------------------------------------------------------
__global__ __launch_bounds__(256) void mark_kernel(const float* __restrict__ labels,
                                                   unsigned char* __restrict__ map,
                                                   int n) {
  int i = blockIdx.x * blockDim.x + threadIdx.x;
  if (i >= n) return;
  unsigned v = (unsigned)labels[i];
  map[v] = 1;
}

// ---------------------------------------------------------------------------
// 5) Count distinct = popcount over the bytemap (bytes are 0x00/0x01).
// ---------------------------------------------------------------------------
__global__ __launch_bounds__(256) void count_kernel(const uint4* __restrict__ map,
                                                    int n16, unsigned* __restrict__ cnt) {
  unsigned s = 0;
  for (int i = blockIdx.x * blockDim.x + threadIdx.x; i < n16;
       i += gridDim.x * blockDim.x) {
    uint4 v = map[i];
    s += __popc(v.x) + __popc(v.y) + __popc(v.z) + __popc(v.w);
  }
  for (int off = 16; off > 0; off >>= 1) s += __shfl_xor(s, off, 32);
  __shared__ unsigned ls[8];
  int wave = threadIdx.x >> 5, lane = threadIdx.x & 31;
  if (lane == 0) ls[wave] = s;
  __syncthreads();
  if (threadIdx.x == 0) {
    unsigned tot = 0;
    for (int w = 0; w < 8; ++w) tot += ls[w];
    atomicAdd(cnt, tot);
  }
}

// ---------------------------------------------------------------------------
// 6) Dice sums per image via V_WMMA_F32_16X16X4_F32. With B = all-ones,
//    D = A*1 + C sums every element of A into each accumulator column —
//    layout-independent, so arbitrary 64-element chunks can be fed as A.
//    One block (32 waves) per image; EXEC is all-ones in the wmma loop.
// ---------------------------------------------------------------------------
__global__ __launch_bounds__(1024) void dice_sums_kernel(const float* __restrict__ p,
                                                         const float* __restrict__ t,
                                                         float* __restrict__ sums) {
  const int img  = blockIdx.x;
  const int tid  = threadIdx.x;
  const int wave = tid >> 5, lane = tid & 31;
  const float* __restrict__ pb = p + (size_t)img * IMG_N;
  const float* __restrict__ tb = t + (size_t)img * IMG_N;

  v8f accN = {};  // sums of p*t
  v8f accD = {};  // sums of p^2 + t^2
  v2f ones; ones[0] = 1.0f; ones[1] = 1.0f;

  // 32 waves * 64 elements per wmma step; IMG_N divides evenly (128 steps).
  for (int base = wave * 64; base < IMG_N; base += 32 * 64) {
    int i = base + lane * 2;
    float2 pv = *(const float2*)(pb + i);
    float2 tv = *(const float2*)(tb + i);
    v2f an; an[0] = pv.x * tv.x;              an[1] = pv.y * tv.y;
    v2f ad; ad[0] = pv.x * pv.x + tv.x * tv.x; ad[1] = pv.y * pv.y + tv.y * tv.y;
    accN = __builtin_amdgcn_wmma_f32_16x16x4_f32(false, an, false, ones,
                                                 (short)0, accN, false, false);
    accD = __builtin_amdgcn_wmma_f32_16x16x4_f32(false, ad, false, ones,
                                                 (short)0, accD, false, false);
  }

  // Column n total = (lane n VGPR fold) + (lane n^16 VGPR fold).
  float sN = 0.0f, sD = 0.0f;
#pragma unroll
  for (int k = 0; k < 8; ++k) { sN += accN[k]; sD += accD[k]; }
  sN += __shfl_xor(sN, 16, 32);
  sD += __shfl_xor(sD, 16, 32);

  __shared__ float redN[32], redD[32];
  if (lane == 0) { redN[wave] = sN; redD[wave] = sD; }
  __syncthreads();
  if (tid == 0) {
    float a = 0.0f, b = 0.0f;
    for (int w = 0; w < 32; ++w) { a += redN[w]; b += redD[w]; }
    sums[img * 2 + 0] = a;
    sums[img * 2 + 1] = b;
  }
}

// ---------------------------------------------------------------------------
// 7) Scalar combine: mean dice loss * blob penalty.
// ---------------------------------------------------------------------------
__global__ void finalize_kernel(const float* __restrict__ sums,
                                const unsigned* __restrict__ cnts,
                                float* __restrict__ out) {
  float mean = 0.0f;
  for (int b = 0; b < BATCH; ++b) {
    float num = sums[b * 2 + 0] + 1.0f;  // SMOOTH
    float den = sums[b * 2 + 1] + 1.0f;
    mean += 1.0f - num / den;
  }
  mean *= (1.0f / (float)BATCH);
  float nl = (float)cnts[0] - 1.0f;  // count_unique(labels) - 1
  float nt = (float)cnts[1];         // count_unique(target_number), per reference
  float pen = sqrtf(nl / nt);
  if (!isfinite(pen)) pen = (float)BATCH;
  pen = fminf(fmaxf(pen, 1.0f), (float)BATCH);
  out[0] = mean * pen;
}

// ---------------------------------------------------------------------------
// Launch sequence (all on `stream`; graph-capture safe).
// ---------------------------------------------------------------------------
extern "C" void kernel_launch(void* const* d_in, const int* in_sizes, int n_in,
                              void* d_out, int out_size, void* d_ws, size_t ws_size,
                              hipStream_t stream) {
  (void)n_in; (void)out_size; (void)ws_size;
  const float* predict = (const float*)d_in[0];
  const float* target  = (const float*)d_in[1];
  float* out = (float*)d_out;
  char* ws = (char*)d_ws;
  const int n = in_sizes[0];  // 4,194,304

  float*         bufA = (float*)(ws + OFF_BUFA);
  float*         bufB = (float*)(ws + OFF_BUFB);
  unsigned char* mapP = (unsigned char*)(ws + OFF_MAPP);
  unsigned char* mapT = (unsigned char*)(ws + OFF_MAPT);
  unsigned*      scal = (unsigned*)(ws + OFF_SCAL);  // keyP,keyT,cntP,cntT
  float*         sums = (float*)(ws + OFF_SUMS);

  // Zero maps + 4 scalar words (contiguous region starting at OFF_MAPP).
  const int zwords = (int)((2 * MAP_BYTES) / 4) + 4;
  init_ws_kernel<<<1024, 256, 0, stream>>>((unsigned*)(ws + OFF_MAPP), zwords);

  max_kernel<<<512, 256, 0, stream>>>(predict, target, &scal[0], n);

  const dim3 cgrid(IMG_W / TILE, IMG_H / TILE, BATCH);
  const int gblocks = n / 256;

  // ---- predict labels ----
  seed_kernel<<<gblocks, 256, 0, stream>>>(predict, scal, 0, bufA, n);
  for (int j = 0; j < CCL_LAUNCHES; ++j) {
    const float* src = (j & 1) ? bufB : bufA;
    float*       dst = (j & 1) ? bufA : bufB;
    ccl_kernel<<<cgrid, 256, 0, stream>>>(src, dst);
  }
  mark_kernel<<<gblocks, 256, 0, stream>>>(bufA, mapP, n);

  // ---- target labels (reuse buffers) ----
  seed_kernel<<<gblocks, 256, 0, stream>>>(target, scal, 1, bufA, n);
  for (int j = 0; j < CCL_LAUNCHES; ++j) {
    const float* src = (j & 1) ? bufB : bufA;
    float*       dst = (j & 1) ? bufA : bufB;
    ccl_kernel<<<cgrid, 256, 0, stream>>>(src, dst);
  }
  mark_kernel<<<gblocks, 256, 0, stream>>>(bufA, mapT, n);

  count_kernel<<<256, 256, 0, stream>>>((const uint4*)mapP, (int)(MAP_BYTES / 16), &scal[2]);
  count_kernel<<<256, 256, 0, stream>>>((const uint4*)mapT, (int)(MAP_BYTES / 16), &scal[3]);

  dice_sums_kernel<<<BATCH, 1024, 0, stream>>>(predict, target, sums);
  finalize_kernel<<<1, 1, 0, stream>>>(sums, &scal[2], out);
}